// NonlienarFeatures_47931835023760
// MI455X (gfx1250) — compile-verified
//
#include <hip/hip_runtime.h>
#include <stdint.h>

// ---------------------------------------------------------------------------
// Problem constants (from reference: L_WIN=50, RHO=1.0, NMODES=2, B=4, S=4096)
// ---------------------------------------------------------------------------
#define BATCH   4
#define SEQ     4096
#define NM      2
#define HALF    25          // L_WIN/2
#define KPAIRS  449         // #(m,n) with |m*n| <= 25
#define TILE    64          // s-values per block
#define WLEN    (TILE + 4 * HALF + 1)   // 165: window [s0-50, s0+TILE+50]
#define NBLOCKS (BATCH * (SEQ / TILE))  // 256

// ---------------------------------------------------------------------------
// Compile-time (m,n) index table, identical enumeration order to _index()
// ---------------------------------------------------------------------------
struct IdxTab { short m[KPAIRS]; short n[KPAIRS]; };

static constexpr int count_pairs() {
  int c = 0;
  for (int m = -HALF; m <= HALF; ++m)
    for (int n = -HALF; n <= HALF; ++n) {
      int p = m * n; if (p < 0) p = -p;
      if (p <= HALF) ++c;
    }
  return c;
}
static_assert(count_pairs() == KPAIRS, "index table size mismatch");

static constexpr IdxTab make_tab() {
  IdxTab t{};
  int c = 0;
  for (int m = -HALF; m <= HALF; ++m)
    for (int n = -HALF; n <= HALF; ++n) {
      int p = m * n; if (p < 0) p = -p;
      if (p <= HALF) { t.m[c] = (short)m; t.n[c] = (short)n; ++c; }
    }
  return t;
}
__constant__ IdxTab c_tab = make_tab();

// ---------------------------------------------------------------------------
// CDNA5 async global -> LDS copy (ASYNCcnt path, CDNA5 ISA)
// Builtin prototype (from hipcc diagnostic): param0 = int4 addrspace(1)*,
// so cast through properly typed, address-space-qualified vector pointers.
// ---------------------------------------------------------------------------
typedef int v4i __attribute__((ext_vector_type(4)));
typedef __attribute__((address_space(1))) v4i* g4p_t;   // global int4*
typedef __attribute__((address_space(3))) v4i* l4p_t;   // LDS int4*

__device__ __forceinline__ void async_copy_b128(const void* g, void* l) {
#if __has_builtin(__builtin_amdgcn_global_load_async_to_lds_b128)
  __builtin_amdgcn_global_load_async_to_lds_b128(
      (g4p_t)(const_cast<void*>(g)),   // addrspacecast generic -> global(1)
      (l4p_t)(l),                      // addrspacecast generic -> LDS(3)
      0, 0);
#else
  // GV mode: vdst = LDS byte address, vaddr = 64-bit global address
  uint32_t loff = (uint32_t)(uintptr_t)(l4p_t)(l);
  uint64_t ga   = (uint64_t)(uintptr_t)g;
  asm volatile("global_load_async_to_lds_b128 %0, %1, off"
               :: "v"(loff), "v"(ga) : "memory");
#endif
}

__device__ __forceinline__ void wait_async_done() {
#if __has_builtin(__builtin_amdgcn_s_wait_asynccnt)
  __builtin_amdgcn_s_wait_asynccnt(0);
#else
  asm volatile("s_wait_asynccnt 0" ::: "memory");
#endif
}

// ---------------------------------------------------------------------------
// Main kernel: one block = (batch b, 64 consecutive s). Stage shifted windows
// of U/V/W (both modes packed as one float4 per s) into LDS via async DMA,
// then stream 2*449 complex outputs per s with coalesced b64 stores.
// out[b][s][mode][k] (complex64), i.e. float2 at ((b*S+s)*2+mode)*K + k.
// ---------------------------------------------------------------------------
__global__ __launch_bounds__(256)
void nonlinear_features_kernel(const float4* __restrict__ U4,
                               const float4* __restrict__ V4,
                               const float4* __restrict__ W4,
                               float2* __restrict__ out) {
  __shared__ float4 sU[WLEN];
  __shared__ float4 sV[WLEN];
  __shared__ float4 sW[WLEN];
  __shared__ short  sM[KPAIRS];
  __shared__ short  sN[KPAIRS];

  const int b   = blockIdx.x / (SEQ / TILE);
  const int s0  = (blockIdx.x % (SEQ / TILE)) * TILE;
  const int tid = threadIdx.x;

  const float4* Ub = U4 + (size_t)b * SEQ;   // one float4 per s = both modes
  const float4* Vb = V4 + (size_t)b * SEQ;
  const float4* Wb = W4 + (size_t)b * SEQ;

  // Stage window [s0-50, s0+TILE+50] with wrap-around handled per lane.
  const int base = s0 - 2 * HALF;
  for (int j = tid; j < WLEN; j += 256) {
    const int g = (base + j + SEQ) & (SEQ - 1);
    async_copy_b128(&Ub[g], &sU[j]);
    async_copy_b128(&Vb[g], &sV[j]);
    async_copy_b128(&Wb[g], &sW[j]);
  }
  // (m,n) table into LDS (plain DS stores).
  for (int j = tid; j < KPAIRS; j += 256) {
    sM[j] = c_tab.m[j];
    sN[j] = c_tab.n[j];
  }
  wait_async_done();
  __syncthreads();

  const int wave = tid >> 5;
  const int lane = tid & 31;

  for (int si = wave; si < TILE; si += 8) {
    const int s  = s0 + si;
    const int ls = si + 2 * HALF;               // window index of s
    float2* o0 = out + (size_t)(b * SEQ + s) * NM * KPAIRS;  // mode 0 row
    float2* o1 = o0 + KPAIRS;                                // mode 1 row

    for (int k = lane; k < KPAIRS; k += 32) {
      const int m = sM[k];
      const int n = sN[k];
      const float4 u = sU[ls - m];          // U[(s-m)%S],   modes 0,1
      const float4 w = sW[ls - m - n];      // W[(s-m-n)%S], modes 0,1
      const float4 v = sV[ls - n];          // V[(s-n)%S],   modes 0,1

      // Asum = U0*conj(W0) + U1*conj(W1)
      const float ar = u.x * w.x + u.y * w.y + u.z * w.z + u.w * w.w;
      const float ai = u.y * w.x - u.x * w.y + u.w * w.z - u.z * w.w;

      // feat_mode = Asum * V_mode
      o0[k] = make_float2(ar * v.x - ai * v.y, ar * v.y + ai * v.x);
      o1[k] = make_float2(ar * v.z - ai * v.w, ar * v.w + ai * v.z);
    }
  }
}

// ---------------------------------------------------------------------------
// Launch. Inputs (setup_inputs order): U, V, W — each (4,4096,2) complex64.
// Output: (4,4096,2,449) complex64 interpreted as float2 stream.
// ---------------------------------------------------------------------------
extern "C" void kernel_launch(void* const* d_in, const int* in_sizes, int n_in,
                              void* d_out, int out_size, void* d_ws, size_t ws_size,
                              hipStream_t stream) {
  (void)in_sizes; (void)n_in; (void)out_size; (void)d_ws; (void)ws_size;
  nonlinear_features_kernel<<<NBLOCKS, 256, 0, stream>>>(
      (const float4*)d_in[0],
      (const float4*)d_in[1],
      (const float4*)d_in[2],
      (float2*)d_out);
}